// MaskedBatchNorm_30253749633578
// MI455X (gfx1250) — compile-verified
//
#include <hip/hip_runtime.h>

typedef float v2f __attribute__((ext_vector_type(2)));
typedef float v4f __attribute__((ext_vector_type(4)));
typedef float v8f __attribute__((ext_vector_type(8)));

#define B_DIM 16
#define N_DIM 8192
#define F_DIM 256
#define ROWS (B_DIM * N_DIM)   // 131072 voxel rows
#define EPS_BN 1e-3f

// ---------------------------------------------------------------------------
// Pass 1: per-block masked partial sums / sums-of-squares, per feature.
// Block = 256 threads. Thread t owns features [4c, 4c+3], c = t & 63, and
// walks rows r = start + (t>>6) + 4*i  (4 row-subgroups per block).
// Wave-level access: 32 lanes * 16B = 512B contiguous per load -> full
// coalescing. Default (RT) temporal hint keeps the 134MB input L2-resident
// for pass 3 (L2 = 192MB on MI455X).
// ---------------------------------------------------------------------------
__global__ void mbn_partials(const float* __restrict__ x,
                             const int* __restrict__ nvalid,
                             float* __restrict__ pSum,
                             float* __restrict__ pSq,
                             int rowsPerBlock) {
  __shared__ int nvs[B_DIM];
  __shared__ float lsum[4][F_DIM];
  __shared__ float lsq[4][F_DIM];

  const int t = threadIdx.x;
  if (t < B_DIM) nvs[t] = nvalid[t];
  __syncthreads();

  const int c   = t & 63;   // feature quad -> features 4c..4c+3
  const int sub = t >> 6;   // row subgroup 0..3
  const int start = blockIdx.x * rowsPerBlock;
  int end = start + rowsPerBlock;
  if (end > ROWS) end = ROWS;

  v4f s = {0.f, 0.f, 0.f, 0.f};
  v4f q = {0.f, 0.f, 0.f, 0.f};

  for (int r = start + sub; r < end; r += 4) {
    const int b = r >> 13;            // N = 8192
    const int n = r & (N_DIM - 1);
    const float m = (n < nvs[b]) ? 1.0f : 0.0f;
    v4f xv = *(const v4f*)(x + (size_t)r * F_DIM + 4 * c);
    s += xv * m;
    q += (xv * xv) * m;
  }

#pragma unroll
  for (int j = 0; j < 4; ++j) {
    lsum[sub][4 * c + j] = s[j];
    lsq [sub][4 * c + j] = q[j];
  }
  __syncthreads();

  // 256 threads: combine the 4 row-subgroups for one feature each (fixed
  // order -> deterministic), write per-block partials to workspace.
  const float ts = lsum[0][t] + lsum[1][t] + lsum[2][t] + lsum[3][t];
  const float tq = lsq [0][t] + lsq [1][t] + lsq [2][t] + lsq [3][t];
  pSum[(size_t)blockIdx.x * F_DIM + t] = ts;
  pSq [(size_t)blockIdx.x * F_DIM + t] = tq;
}

// ---------------------------------------------------------------------------
// Pass 2: final [K][256] -> [256] column reduction via V_WMMA_F32_16X16X4_F32
// with an all-ones A matrix (D[m,n] = sum_k B[k,n]); since we sum over every
// K slot, the exact K->lane/VGPR mapping of B is irrelevant — only the
// column<->lane (lane&15) mapping matters. Then compute mean and
// scale = rsqrt(var+eps)*gamma. 32 waves: (sum|sq) x 16 feature groups.
// K is a multiple of 4 -> the loop is branch-free (EXEC all-1s for WMMA).
// ---------------------------------------------------------------------------
__global__ void mbn_finalize(const float* __restrict__ pSum,
                             const float* __restrict__ pSq,
                             const int* __restrict__ nvalid,
                             const float* __restrict__ gamma,
                             float* __restrict__ meanOut,
                             float* __restrict__ scaleOut,
                             int K) {
  __shared__ float totS[F_DIM];
  __shared__ float totQ[F_DIM];

  const int t    = threadIdx.x;
  const int lane = t & 31;
  const int wave = t >> 5;        // 0..31
  const int arr  = wave & 1;      // 0 = sum, 1 = sumsq
  const int g    = wave >> 1;     // feature group 0..15
  const int f0   = g * 16;

  const float* __restrict__ P = arr ? pSq : pSum;
  const int col = lane & 15;
  const int klo = (lane < 16) ? 0 : 2;   // which K slots this lane-half feeds

  v2f aOnes = {1.0f, 1.0f};
  v8f acc = {0.f, 0.f, 0.f, 0.f, 0.f, 0.f, 0.f, 0.f};

  for (int k = 0; k < K; k += 4) {
    const size_t r0 = (size_t)(k + klo) * F_DIM + f0 + col;
    v2f bv;
    bv.x = P[r0];
    bv.y = P[r0 + F_DIM];
    // D = A(ones) x B + D  ->  every D element in lane L = colsum over K slots
    acc = __builtin_amdgcn_wmma_f32_16x16x4_f32(
        /*neg_a=*/false, aOnes, /*neg_b=*/false, bv,
        /*c_mod=*/(short)0, acc, /*reuse_a=*/false, /*reuse_b=*/false);
  }

  if (lane < 16) {
    if (arr == 0) totS[f0 + lane] = acc[0];
    else          totQ[f0 + lane] = acc[0];
  }
  __syncthreads();

  if (t < F_DIM) {
    float cnt = 0.0f;
    for (int i = 0; i < B_DIM; ++i) cnt += (float)nvalid[i];
    cnt = fmaxf(cnt, 1.0f);
    const float mean = totS[t] / cnt;
    const float var  = totQ[t] / cnt - mean * mean;
    const float inv  = rsqrtf(var + EPS_BN);
    meanOut[t]  = mean;
    scaleOut[t] = inv * gamma[t];
  }
}

// ---------------------------------------------------------------------------
// Pass 3: elementwise normalize + mask. One float4 per thread. Loads are RT
// (hit the L2 populated by pass 1); output is streamed with non-temporal
// stores so the 134MB write does not evict the L2-resident input.
// ---------------------------------------------------------------------------
__global__ void mbn_normalize(const float* __restrict__ x,
                              const int* __restrict__ nvalid,
                              const float* __restrict__ meanP,
                              const float* __restrict__ scaleP,
                              const float* __restrict__ beta,
                              float* __restrict__ out) {
  const size_t tid = (size_t)blockIdx.x * blockDim.x + threadIdx.x;
  const size_t e4  = tid * 4;                 // element index (float4 granular)
  const int r = (int)(e4 >> 8);               // row, F = 256
  const int f = (int)(e4 & 255);
  const int b = r >> 13;
  const int n = r & (N_DIM - 1);
  const bool valid = n < nvalid[b];

  v4f xv = *(const v4f*)(x + e4);
  v4f mv = *(const v4f*)(meanP + f);
  v4f sv = *(const v4f*)(scaleP + f);
  v4f bv = *(const v4f*)(beta + f);

  v4f o = (xv - mv) * sv + bv;
  if (!valid) { o.x = 0.f; o.y = 0.f; o.z = 0.f; o.w = 0.f; }

  __builtin_nontemporal_store(o, (v4f*)(out + e4));
}

// ---------------------------------------------------------------------------
// Host launcher
// inputs: [0] voxel_features f32 [16,8192,256]
//         [1] num_valid_voxels i32 [16]
//         [2] gamma f32 [256]
//         [3] beta  f32 [256]
// output: f32 [16,8192,256]
// ---------------------------------------------------------------------------
extern "C" void kernel_launch(void* const* d_in, const int* in_sizes, int n_in,
                              void* d_out, int out_size, void* d_ws, size_t ws_size,
                              hipStream_t stream) {
  (void)in_sizes; (void)n_in; (void)out_size;

  const float* x     = (const float*)d_in[0];
  const int*   nv    = (const int*)d_in[1];
  const float* gamma = (const float*)d_in[2];
  const float* beta  = (const float*)d_in[3];
  float*       out   = (float*)d_out;
  float*       ws    = (float*)d_ws;

  // Size the pass-1 grid from available workspace:
  // per block: 256 sum + 256 sq floats = 512 floats; params: 512 floats.
  long availFloats = (long)(ws_size / sizeof(float));
  long nb = (availFloats - 512) / 512;
  nb &= ~3L;                 // multiple of 4 -> branch-free WMMA K-loop
  if (nb < 4)   nb = 4;
  if (nb > 512) nb = 512;
  const int NB = (int)nb;
  const int rowsPerBlock = (ROWS + NB - 1) / NB;

  float* pSum  = ws;
  float* pSq   = ws + (size_t)NB * F_DIM;
  float* meanP = ws + (size_t)NB * F_DIM * 2;
  float* scale = meanP + F_DIM;

  mbn_partials<<<NB, 256, 0, stream>>>(x, nv, pSum, pSq, rowsPerBlock);
  mbn_finalize<<<1, 1024, 0, stream>>>(pSum, pSq, nv, gamma, meanP, scale, NB);

  const int totalQuads = ROWS * F_DIM / 4;           // 8,388,608
  mbn_normalize<<<totalQuads / 256, 256, 0, stream>>>(x, nv, meanP, scale, beta, out);
}